// Grid2DPartialPositiver_36773509988519
// MI455X (gfx1250) — compile-verified
//
#include <hip/hip_runtime.h>
#include <stdint.h>

// CDNA5 / gfx1250: wave32, 128-bit VMEM, NT temporal hints for streaming data.
typedef float v4f __attribute__((ext_vector_type(4)));

// One 256x256 plane = 16384 float4 elements. Each block handles 2048 float4
// (256 threads x 8 float4), so 8 blocks per plane; the channel mask is
// wave-uniform per block -> single scalar branch, no divergence.
#define F4_PER_PLANE     16384
#define BLOCKS_PER_PLANE 8
#define ITERS            8
#define THREADS          256

// Explicit relu: guarantee a real v_max_num_f32 against inline-constant 0.
// (The __builtin_fmaxf(x, 0.0f) path lowered to v_max_num_f32 v,v,v — a
// canonicalize with the zero operand dropped — so emit the op directly.
// VOP2 form: src0 may be an inline constant, vsrc1 must be a VGPR.)
__device__ __forceinline__ float relu_f32(float a) {
    float r;
    asm("v_max_num_f32 %0, 0, %1" : "=v"(r) : "v"(a));
    return r;
}

// POW2: channel = plane & cArg (cArg = C-1, single s_and_b32).
// !POW2: channel = plane % cArg (generic fallback).
template <bool POW2>
__global__ __launch_bounds__(THREADS)
void grid2d_partial_positiver_kernel(const float* __restrict__ x,
                                     const uint8_t* __restrict__ posIdx,
                                     float* __restrict__ out,
                                     int cArg) {
    const int sub   = blockIdx.x & (BLOCKS_PER_PLANE - 1);
    const int plane = blockIdx.x >> 3;          // n*C + c
    const int c     = POW2 ? (plane & cArg) : (plane % cArg);

    // Wave-uniform mask: entire block processes one channel's plane chunk.
    const bool applyRelu = (posIdx[c] != 0);

    const long base = (long)plane * F4_PER_PLANE
                    + (long)sub * (THREADS * ITERS)
                    + threadIdx.x;

    const v4f* __restrict__ xin = ((const v4f*)x) + base;
    v4f*       __restrict__ yo  = ((v4f*)out)     + base;

    // Issue all 8 128-bit NT loads first: 8 global_load_b128 th:NT in flight
    // per thread for maximum memory-level parallelism.
    v4f v[ITERS];
#pragma unroll
    for (int i = 0; i < ITERS; ++i)
        v[i] = __builtin_nontemporal_load(xin + i * THREADS);

    if (applyRelu) {
#pragma unroll
        for (int i = 0; i < ITERS; ++i) {
            v[i].x = relu_f32(v[i].x);
            v[i].y = relu_f32(v[i].y);
            v[i].z = relu_f32(v[i].z);
            v[i].w = relu_f32(v[i].w);
        }
    }

    // Streaming 128-bit NT stores (th:TH_STORE_NT): output is never re-read,
    // don't let 512MB of one-touch data evict the 192MB L2.
#pragma unroll
    for (int i = 0; i < ITERS; ++i)
        __builtin_nontemporal_store(v[i], yo + i * THREADS);
}

extern "C" void kernel_launch(void* const* d_in, const int* in_sizes, int n_in,
                              void* d_out, int out_size, void* d_ws, size_t ws_size,
                              hipStream_t stream) {
    const float*   x      = (const float*)d_in[0];
    const uint8_t* posIdx = (const uint8_t*)d_in[1];  // JAX bool -> 1 byte/elem
    float*         out    = (float*)d_out;

    const int total   = in_sizes[0];                  // 16*64*256*256
    const int C       = in_sizes[1];                  // 64
    const int nPlanes = total / (F4_PER_PLANE * 4);   // 1024 planes of 256*256
    const int nBlocks = nPlanes * BLOCKS_PER_PLANE;   // 8192

    if ((C & (C - 1)) == 0) {
        grid2d_partial_positiver_kernel<true>
            <<<dim3(nBlocks), dim3(THREADS), 0, stream>>>(x, posIdx, out, C - 1);
    } else {
        grid2d_partial_positiver_kernel<false>
            <<<dim3(nBlocks), dim3(THREADS), 0, stream>>>(x, posIdx, out, C);
    }
}